// K24_RNNAgent_1R3_18906446037361
// MI455X (gfx1250) — compile-verified
//
#include <hip/hip_runtime.h>
#include <math.h>

// ---------------------------------------------------------------------------
// K24 RNNAgent: per-agent 2:4 soft-masked MLP + GRU, fp32 WMMA on gfx1250.
//   Phase 1: masked weights wm[a] = w * (softmax(alpha/5) @ PATTERNS)
//   Phase 2: counting-sort rows by agent (segments 32-aligned) -> rowMap
//   Phase 3: WMMA f32 16x16x4 GEMMs, M-blocked x2 to halve B-fragment traffic
//   Phase 4: scatter q[8192,16], h[8192,512] back to original row order
// ---------------------------------------------------------------------------

typedef __attribute__((ext_vector_type(2))) float v2f;
typedef __attribute__((ext_vector_type(8))) float v8f;

static constexpr int N_ROWS = 8192;     // B*A = 1024*8
static constexpr int NPAD   = 8448;     // 8192 + 8*32 padding headroom (32-aligned segs)
static constexpr int MT32   = NPAD / 32;
static constexpr int E_DIM  = 64;
static constexpr int H_DIM  = 512;
static constexpr int NA_DIM = 16;
static constexpr int AGENTS = 8;

// ----------------------------- setup kernels -------------------------------

__global__ void init_kernel(int* counts, int* cursors, int* rowMap, int npad) {
    int i = blockIdx.x * blockDim.x + threadIdx.x;
    if (i < 8) { counts[i] = 0; cursors[i] = 0; }
    if (i < npad) rowMap[i] = -1;
}

__global__ void hist_kernel(const int* __restrict__ ids, int* counts, int n) {
    int i = blockIdx.x * blockDim.x + threadIdx.x;
    if (i < n) atomicAdd(&counts[ids[i]], 1);
}

__global__ void scan_kernel(const int* __restrict__ counts, int* padStart) {
    if (threadIdx.x == 0 && blockIdx.x == 0) {
        int acc = 0;
        for (int a = 0; a < 8; ++a) {
            padStart[a] = acc;
            acc += (counts[a] + 31) & ~31;   // 32-align each agent segment
        }
        padStart[8] = acc;
    }
}

__global__ void perm_kernel(const int* __restrict__ ids, const int* __restrict__ padStart,
                            int* cursors, int* rowMap, int n) {
    int i = blockIdx.x * blockDim.x + threadIdx.x;
    if (i < n) {
        int a = ids[i];
        int pos = padStart[a] + atomicAdd(&cursors[a], 1);
        rowMap[pos] = i;
    }
}

__global__ void gather_kernel(const int* __restrict__ rowMap,
                              const float* __restrict__ x, const float* __restrict__ h,
                              float* __restrict__ xg, float* __restrict__ h0g) {
    int p = blockIdx.x;
    int t = threadIdx.x;                 // 128 threads
    int n = rowMap[p];
    if (t < E_DIM)
        xg[(size_t)p * E_DIM + t] = (n >= 0) ? x[(size_t)n * E_DIM + t] : 0.f;
    for (int c = t; c < H_DIM; c += 128)
        h0g[(size_t)p * H_DIM + c] = (n >= 0) ? h[(size_t)n * H_DIM + c] : 0.f;
}

// masked weight: wm[a,o,g*4+k] = w[o,g*4+k] * sum_p softmax(alpha/5)[p]*PAT[p][k]
__global__ void maskw_kernel(const float* __restrict__ alpha, const float* __restrict__ w,
                             float* __restrict__ wm, int O, int G, int total) {
    int idx = blockIdx.x * blockDim.x + threadIdx.x;
    if (idx >= total) return;
    int g = idx % G;
    int o = (idx / G) % O;
    int a = idx / (G * O);
    const float* al = alpha + (size_t)idx * 6;
    float mx = al[0];
    #pragma unroll
    for (int p = 1; p < 6; ++p) mx = fmaxf(mx, al[p]);
    float e[6], s = 0.f;
    #pragma unroll
    for (int p = 0; p < 6; ++p) { e[p] = __expf((al[p] - mx) * 0.2f); s += e[p]; }
    float inv = 1.f / s;
    // PATTERNS: [1100][1010][1001][0110][0101][0011]
    float k0 = (e[0] + e[1] + e[2]) * inv;
    float k1 = (e[0] + e[3] + e[4]) * inv;
    float k2 = (e[1] + e[3] + e[5]) * inv;
    float k3 = (e[2] + e[4] + e[5]) * inv;
    size_t wi = (size_t)o * (G * 4) + (size_t)g * 4;
    size_t wo = (size_t)a * O * (size_t)(G * 4) + wi;
    wm[wo + 0] = w[wi + 0] * k0;
    wm[wo + 1] = w[wi + 1] * k1;
    wm[wo + 2] = w[wi + 2] * k2;
    wm[wo + 3] = w[wi + 3] * k3;
}

// --------------------------- WMMA f32 GEMMs --------------------------------
// One wave (32 lanes) computes a (16*MB) x (16*NB) tile. Agent is uniform per
// tile because segments are 32-aligned. A: 16x4 fp32 fragment = v2f per lane
// (lane&15 = M row, lane>>4 selects K pair {0,1} vs {2,3}); B symmetric with
// N = lane&15; C/D: v8f, element v -> M = v + 8*(lane>>4), N = lane&15.

template<int MB, int NB, bool RELU>
__global__ void __launch_bounds__(32)
gemm_agent(const float* __restrict__ X, int K,
           const float* __restrict__ WM, const float* __restrict__ bias, int O,
           const int* __restrict__ padStart, float* __restrict__ Y) {
    const int m0 = blockIdx.x * (16 * MB);
    const int n0 = blockIdx.y * (16 * NB);
    int a = 0;
    #pragma unroll
    for (int t = 0; t < 7; ++t)
        if (m0 >= padStart[t + 1]) a = t + 1;

    const int lane = threadIdx.x;
    const int row  = lane & 15;
    const int grp  = lane >> 4;            // 0 or 1 (K pair select)
    const float* Arow[MB];
    #pragma unroll
    for (int mi = 0; mi < MB; ++mi)
        Arow[mi] = X + (size_t)(m0 + mi * 16 + row) * K + 2 * grp;
    const float* Wb = WM + (size_t)a * O * K;
    const float* Wr[NB];
    #pragma unroll
    for (int j = 0; j < NB; ++j)
        Wr[j] = Wb + (size_t)(n0 + j * 16 + row) * K + 2 * grp;

    v8f acc[MB][NB] = {};
    for (int k0 = 0; k0 < K; k0 += 4) {
        v2f av[MB];
        #pragma unroll
        for (int mi = 0; mi < MB; ++mi) av[mi] = *(const v2f*)(Arow[mi] + k0);
        #pragma unroll
        for (int j = 0; j < NB; ++j) {
            v2f bv = *(const v2f*)(Wr[j] + k0);
            #pragma unroll
            for (int mi = 0; mi < MB; ++mi)
                acc[mi][j] = __builtin_amdgcn_wmma_f32_16x16x4_f32(
                    false, av[mi], false, bv, (short)0, acc[mi][j], false, false);
        }
    }

    #pragma unroll
    for (int j = 0; j < NB; ++j) {
        int col = n0 + j * 16 + row;
        float bv = bias[col];
        #pragma unroll
        for (int mi = 0; mi < MB; ++mi) {
            #pragma unroll
            for (int v = 0; v < 8; ++v) {
                int m = m0 + mi * 16 + v + grp * 8;
                float y = acc[mi][j][v] + bv;
                if (RELU) y = fmaxf(y, 0.f);
                Y[(size_t)m * O + col] = y;
            }
        }
    }
}

// Fused GRU: one wave per 32x16 tile of H (M-blocked x2); six gate accumulators
// per M sub-tile (ir,hr,iz,hz,in,hn), gate math in-register, writes new hidden.
__global__ void __launch_bounds__(32)
gru_kernel(const float* __restrict__ x1, const float* __restrict__ h0,
           const float* __restrict__ w_ih, const float* __restrict__ w_hh,
           const float* __restrict__ b_ih, const float* __restrict__ b_hh,
           float* __restrict__ H) {
    constexpr int MB = 2;
    const int m0 = blockIdx.x * (16 * MB);
    const int c0 = blockIdx.y * 16;
    const int lane = threadIdx.x;
    const int row  = lane & 15;
    const int grp  = lane >> 4;

    const float* Ax[MB];
    const float* Ah[MB];
    #pragma unroll
    for (int mi = 0; mi < MB; ++mi) {
        Ax[mi] = x1 + (size_t)(m0 + mi * 16 + row) * H_DIM + 2 * grp;
        Ah[mi] = h0 + (size_t)(m0 + mi * 16 + row) * H_DIM + 2 * grp;
    }
    const int wn = c0 + row;
    const float* Wir = w_ih + (size_t)wn * H_DIM + 2 * grp;
    const float* Wiz = w_ih + (size_t)(wn + 512)  * H_DIM + 2 * grp;
    const float* Win = w_ih + (size_t)(wn + 1024) * H_DIM + 2 * grp;
    const float* Whr = w_hh + (size_t)wn * H_DIM + 2 * grp;
    const float* Whz = w_hh + (size_t)(wn + 512)  * H_DIM + 2 * grp;
    const float* Whn = w_hh + (size_t)(wn + 1024) * H_DIM + 2 * grp;

    v8f air[MB] = {}, ahr[MB] = {}, aiz[MB] = {}, ahz[MB] = {}, ain[MB] = {}, ahn[MB] = {};
    for (int k0 = 0; k0 < H_DIM; k0 += 4) {
        v2f ax[MB], ah[MB];
        #pragma unroll
        for (int mi = 0; mi < MB; ++mi) {
            ax[mi] = *(const v2f*)(Ax[mi] + k0);
            ah[mi] = *(const v2f*)(Ah[mi] + k0);
        }
        v2f b;
        b = *(const v2f*)(Wir + k0);
        #pragma unroll
        for (int mi = 0; mi < MB; ++mi)
            air[mi] = __builtin_amdgcn_wmma_f32_16x16x4_f32(false, ax[mi], false, b, (short)0, air[mi], false, false);
        b = *(const v2f*)(Wiz + k0);
        #pragma unroll
        for (int mi = 0; mi < MB; ++mi)
            aiz[mi] = __builtin_amdgcn_wmma_f32_16x16x4_f32(false, ax[mi], false, b, (short)0, aiz[mi], false, false);
        b = *(const v2f*)(Win + k0);
        #pragma unroll
        for (int mi = 0; mi < MB; ++mi)
            ain[mi] = __builtin_amdgcn_wmma_f32_16x16x4_f32(false, ax[mi], false, b, (short)0, ain[mi], false, false);
        b = *(const v2f*)(Whr + k0);
        #pragma unroll
        for (int mi = 0; mi < MB; ++mi)
            ahr[mi] = __builtin_amdgcn_wmma_f32_16x16x4_f32(false, ah[mi], false, b, (short)0, ahr[mi], false, false);
        b = *(const v2f*)(Whz + k0);
        #pragma unroll
        for (int mi = 0; mi < MB; ++mi)
            ahz[mi] = __builtin_amdgcn_wmma_f32_16x16x4_f32(false, ah[mi], false, b, (short)0, ahz[mi], false, false);
        b = *(const v2f*)(Whn + k0);
        #pragma unroll
        for (int mi = 0; mi < MB; ++mi)
            ahn[mi] = __builtin_amdgcn_wmma_f32_16x16x4_f32(false, ah[mi], false, b, (short)0, ahn[mi], false, false);
    }

    const int col = c0 + row;
    float bir = b_ih[col], biz = b_ih[col + 512], bin = b_ih[col + 1024];
    float bhr = b_hh[col], bhz = b_hh[col + 512], bhn = b_hh[col + 1024];
    #pragma unroll
    for (int mi = 0; mi < MB; ++mi) {
        #pragma unroll
        for (int v = 0; v < 8; ++v) {
            int m = m0 + mi * 16 + v + grp * 8;
            float r = 1.f / (1.f + expf(-(air[mi][v] + bir + ahr[mi][v] + bhr)));
            float z = 1.f / (1.f + expf(-(aiz[mi][v] + biz + ahz[mi][v] + bhz)));
            float n = tanhf(ain[mi][v] + bin + r * (ahn[mi][v] + bhn));
            float hp = h0[(size_t)m * H_DIM + col];
            H[(size_t)m * H_DIM + col] = (1.f - z) * n + z * hp;
        }
    }
}

__global__ void scatter_kernel(const int* __restrict__ rowMap,
                               const float* __restrict__ q4, const float* __restrict__ hbuf,
                               float* __restrict__ outq, float* __restrict__ outh) {
    int p = blockIdx.x;
    int n = rowMap[p];
    if (n < 0) return;
    int t = threadIdx.x;                 // 128 threads
    if (t < NA_DIM) outq[(size_t)n * NA_DIM + t] = q4[(size_t)p * NA_DIM + t];
    for (int c = t; c < H_DIM; c += 128)
        outh[(size_t)n * H_DIM + c] = hbuf[(size_t)p * H_DIM + c];
}

// ------------------------------- launcher ----------------------------------

extern "C" void kernel_launch(void* const* d_in, const int* in_sizes, int n_in,
                              void* d_out, int out_size, void* d_ws, size_t ws_size,
                              hipStream_t stream) {
    const float* inputs   = (const float*)d_in[0];
    const float* hidden   = (const float*)d_in[1];
    const int*   ids      = (const int*)  d_in[2];
    const float* fc1_w    = (const float*)d_in[3];
    const float* fc1_b    = (const float*)d_in[4];
    const float* fc1_al   = (const float*)d_in[5];
    const float* gru_wih  = (const float*)d_in[6];
    const float* gru_whh  = (const float*)d_in[7];
    const float* gru_bih  = (const float*)d_in[8];
    const float* gru_bhh  = (const float*)d_in[9];
    const float* fc2_w    = (const float*)d_in[10];
    const float* fc2_b    = (const float*)d_in[11];
    const float* fc2_al   = (const float*)d_in[12];
    const float* fc3_w    = (const float*)d_in[13];
    const float* fc3_b    = (const float*)d_in[14];
    const float* fc3_al   = (const float*)d_in[15];
    const float* fc4_w    = (const float*)d_in[16];
    const float* fc4_b    = (const float*)d_in[17];
    const float* fc4_al   = (const float*)d_in[18];

    // workspace layout
    int* iw       = (int*)d_ws;
    int* counts   = iw;          // 8
    int* cursors  = iw + 8;      // 8
    int* padStart = iw + 16;     // 9 (room for 16)
    int* rowMap   = iw + 32;     // NPAD
    float* fb  = (float*)d_ws + 8704;                   // 34816-byte offset (past ints)
    float* wm1 = fb;                                    // 8*512*64
    float* wm2 = wm1 + (size_t)AGENTS * 512 * 64;       // 8*512*512
    float* wm3 = wm2 + (size_t)AGENTS * 512 * 512;
    float* wm4 = wm3 + (size_t)AGENTS * 512 * 512;      // 8*16*512
    float* xg   = wm4 + (size_t)AGENTS * 16 * 512;      // NPAD*64
    float* bufA = xg   + (size_t)NPAD * E_DIM;          // x1, later q2
    float* bufB = bufA + (size_t)NPAD * H_DIM;          // h0 gathered, later q3
    float* bufH = bufB + (size_t)NPAD * H_DIM;          // new hidden
    float* q4   = bufH + (size_t)NPAD * H_DIM;          // NPAD*16
    (void)in_sizes; (void)n_in; (void)out_size; (void)ws_size;

    // agent bucketing (32-aligned segments)
    init_kernel<<<(NPAD + 255) / 256, 256, 0, stream>>>(counts, cursors, rowMap, NPAD);
    hist_kernel<<<N_ROWS / 256, 256, 0, stream>>>(ids, counts, N_ROWS);
    scan_kernel<<<1, 32, 0, stream>>>(counts, padStart);
    perm_kernel<<<N_ROWS / 256, 256, 0, stream>>>(ids, padStart, cursors, rowMap, N_ROWS);
    gather_kernel<<<NPAD, 128, 0, stream>>>(rowMap, inputs, hidden, xg, bufB);

    // masked weights
    maskw_kernel<<<(8 * 512 * 16 + 255) / 256, 256, 0, stream>>>(fc1_al, fc1_w, wm1, 512, 16, 8 * 512 * 16);
    maskw_kernel<<<(8 * 512 * 128 + 255) / 256, 256, 0, stream>>>(fc2_al, fc2_w, wm2, 512, 128, 8 * 512 * 128);
    maskw_kernel<<<(8 * 512 * 128 + 255) / 256, 256, 0, stream>>>(fc3_al, fc3_w, wm3, 512, 128, 8 * 512 * 128);
    maskw_kernel<<<(8 * 16 * 128 + 255) / 256, 256, 0, stream>>>(fc4_al, fc4_w, wm4, 16, 128, 8 * 16 * 128);

    // pipeline (MB=2, NB=4 tiles: 32x64 per wave; fc4 is 32x16)
    gemm_agent<2, 4, true ><<<dim3(MT32, H_DIM / 64), 32, 0, stream>>>(xg,   E_DIM, wm1, fc1_b, H_DIM, padStart, bufA);
    gru_kernel<<<dim3(MT32, H_DIM / 16), 32, 0, stream>>>(bufA, bufB, gru_wih, gru_whh, gru_bih, gru_bhh, bufH);
    gemm_agent<2, 4, true ><<<dim3(MT32, H_DIM / 64), 32, 0, stream>>>(bufH, H_DIM, wm2, fc2_b, H_DIM, padStart, bufA);
    gemm_agent<2, 4, true ><<<dim3(MT32, H_DIM / 64), 32, 0, stream>>>(bufA, H_DIM, wm3, fc3_b, H_DIM, padStart, bufB);
    gemm_agent<2, 1, false><<<dim3(MT32, 1),          32, 0, stream>>>(bufB, H_DIM, wm4, fc4_b, NA_DIM, padStart, q4);

    // scatter back to original row order: q first, then h
    scatter_kernel<<<NPAD, 128, 0, stream>>>(rowMap, q4, bufH,
                                             (float*)d_out,
                                             (float*)d_out + (size_t)N_ROWS * NA_DIM);
}